// NeuralMemory_87625922773512
// MI455X (gfx1250) — compile-verified
//
#include <hip/hip_runtime.h>

#define DIMX 512
#define HEADS 8
#define DH 64
#define CHUNKT 64
#define DHID 256
#define NC 64
#define NB 2
#define BH 16
#define EPSV 1e-6f
#define PER_STREAM (DH + DH*DHID + DHID*DH)   // 32832

typedef float v2f __attribute__((ext_vector_type(2)));
typedef float v8f __attribute__((ext_vector_type(8)));

__device__ __forceinline__ v8f wmma4(v2f a, v2f b, v8f c) {
  // V_WMMA_F32_16X16X4_F32 : D = A(16x4) * B(4x16) + C(16x16), fp32
  return __builtin_amdgcn_wmma_f32_16x16x4_f32(false, a, false, b, (short)0, c,
                                               false, false);
}

__device__ __forceinline__ float gelu_f(float x) {
  const float k0 = 0.7978845608028654f, k1 = 0.044715f;
  float u = k0 * (x + k1 * x * x * x);
  return 0.5f * x * (1.f + tanhf(u));
}
__device__ __forceinline__ float gelu_grad(float x) {
  const float k0 = 0.7978845608028654f, k1 = 0.044715f;
  float x2 = x * x;
  float u = k0 * x * (1.f + k1 * x2);
  float t = tanhf(u);
  float sech2 = 1.f - t * t;
  return 0.5f * (1.f + t) + 0.5f * x * sech2 * k0 * (1.f + 3.f * k1 * x2);
}

// ---------------------------------------------------------------------------
// Kernel 1: rmsnorm(seq) -> xhat ; per-token lr ; per-chunk mom/decay gates.
// grid = NB*NC = 128 blocks, 256 threads, dyn LDS = (64*512 + 512) floats.
// ---------------------------------------------------------------------------
__global__ __launch_bounds__(256) void prep_kernel(
    const float* __restrict__ seq, const float* __restrict__ snw,
    const float* __restrict__ Wstep, const float* __restrict__ bstep,
    const float* __restrict__ Wmom, const float* __restrict__ bmom,
    const float* __restrict__ Wdec, const float* __restrict__ bdec,
    float* __restrict__ xhat, float* __restrict__ lr_ws,
    float* __restrict__ m_ws, float* __restrict__ d_ws) {
  extern __shared__ float sm[];
  float* xh = sm;                     // [64][512]
  float* pooled = sm + CHUNKT * DIMX; // [512]
  const int blk = blockIdx.x;
  const int b = blk / NC, j = blk % NC;
  const int tid = threadIdx.x, lane = tid & 31, wave = tid >> 5;
  const float* srow = seq + ((size_t)b * 4096 + (size_t)j * CHUNKT) * DIMX;

  for (int it = 0; it < 8; ++it) {
    int t = wave * 8 + it;
    const float* sr = srow + t * DIMX;
    float ss = 0.f;
    for (int c = lane; c < DIMX; c += 32) { float x = sr[c]; ss += x * x; }
    for (int o = 16; o > 0; o >>= 1) ss += __shfl_xor(ss, o, 32);
    float rstd = rsqrtf(ss / DIMX + EPSV);
    for (int c = lane; c < DIMX; c += 32) xh[t * DIMX + c] = sr[c] * rstd * snw[c];
  }
  __syncthreads();

  float* xg = xhat + ((size_t)b * 4096 + (size_t)j * CHUNKT) * DIMX;
  for (int i = tid; i < CHUNKT * DIMX; i += 256) xg[i] = xh[i];
  for (int c = tid; c < DIMX; c += 256) {
    float s = 0.f;
    for (int t = 0; t < CHUNKT; ++t) s += xh[t * DIMX + c];
    pooled[c] = s * (1.f / CHUNKT);
  }
  __syncthreads();

  // per-token lr: sigmoid(xh @ Wstep + bstep)
  for (int p = tid; p < CHUNKT * HEADS; p += 256) {
    int t = p / HEADS, h = p % HEADS;
    float acc = bstep[h];
    for (int c = 0; c < DIMX; ++c) acc += xh[t * DIMX + c] * Wstep[c * HEADS + h];
    float lr = 1.f / (1.f + __expf(-acc));
    int s_idx = b * HEADS + h;
    lr_ws[((size_t)s_idx * NC + j) * CHUNKT + t] = lr;
  }
  // per-chunk momentum / decay gates
  if (tid < 2 * HEADS) {
    int h = tid % HEADS, which = tid / HEADS;
    const float* W = which ? Wdec : Wmom;
    float acc = which ? bdec[h] : bmom[h];
    for (int c = 0; c < DIMX; ++c) acc += pooled[c] * W[c * HEADS + h];
    float g = 1.f / (1.f + __expf(-acc));
    int s_idx = b * HEADS + h;
    (which ? d_ws : m_ws)[s_idx * NC + j] = g;
  }
}

// ---------------------------------------------------------------------------
// Kernel 2: k = xhat@Wk, v = xhat@Wv via WMMA f32. One 64x64 output tile per
// workgroup: grid = (128 chunks, 8 heads, 2 matrices). K=512 blocked by 64.
// ---------------------------------------------------------------------------
__global__ __launch_bounds__(256) void kv_gemm_kernel(
    const float* __restrict__ xhat, const float* __restrict__ Wk,
    const float* __restrict__ Wv, float* __restrict__ k_ws,
    float* __restrict__ v_ws) {
  extern __shared__ float sm[];
  float* As = sm;            // [64][66]
  float* Bs = sm + 64 * 66;  // [64][66]
  const int blk = blockIdx.x;
  const int b = blk / NC, j = blk % NC;
  const int h = blockIdx.y, kv = blockIdx.z;
  const float* W = kv ? Wv : Wk;
  float* outp = kv ? v_ws : k_ws;
  const int tid = threadIdx.x, lane = tid & 31, wave = tid >> 5;
  const float* xrow = xhat + ((size_t)b * 4096 + (size_t)j * CHUNKT) * DIMX;

  v8f acc[2] = {v8f{}, v8f{}};
  int tmi[2], tni[2];
  for (int q = 0; q < 2; ++q) { int tI = wave * 2 + q; tmi[q] = tI >> 2; tni[q] = tI & 3; }

  for (int kk = 0; kk < DIMX; kk += 64) {
    __syncthreads();
    for (int i = tid; i < 64 * 16; i += 256) {
      int r = i / 16, c4 = (i % 16) * 4;
      float4 s4 = *(const float4*)&xrow[r * DIMX + kk + c4];
      float* dst = &As[r * 66 + c4];
      dst[0] = s4.x; dst[1] = s4.y; dst[2] = s4.z; dst[3] = s4.w;
    }
    for (int i = tid; i < 64 * 16; i += 256) {
      int r = i / 16, c4 = (i % 16) * 4;
      float4 s4 = *(const float4*)&W[(size_t)(kk + r) * DIMX + h * 64 + c4];
      float* dst = &Bs[r * 66 + c4];
      dst[0] = s4.x; dst[1] = s4.y; dst[2] = s4.z; dst[3] = s4.w;
    }
    __syncthreads();
    for (int k4 = 0; k4 < 64; k4 += 4) {
      int kb = k4 + ((lane >> 4) << 1);
      for (int q = 0; q < 2; ++q) {
        int ar = tmi[q] * 16 + (lane & 15);
        int bc = tni[q] * 16 + (lane & 15);
        v2f a = *(v2f*)&As[ar * 66 + kb];
        v2f bb; bb.x = Bs[kb * 66 + bc]; bb.y = Bs[(kb + 1) * 66 + bc];
        acc[q] = wmma4(a, bb, acc[q]);
      }
    }
  }
  int s_idx = b * HEADS + h;
  float* obase = outp + ((size_t)s_idx * NC + j) * CHUNKT * DH;
  for (int q = 0; q < 2; ++q)
    for (int i = 0; i < 8; ++i) {
      int t = tmi[q] * 16 + i + ((lane >> 4) << 3);
      int d = tni[q] * 16 + (lane & 15);
      obase[t * DH + d] = acc[q][i];
    }
}

// ---------------------------------------------------------------------------
// Kernel 3: collapse the two scalar-gated scans into per-chunk coefficients.
// final = A*w0 - sum_j c_j * grad_j. 16 blocks x 64 threads.
// ---------------------------------------------------------------------------
__global__ void coeff_kernel(const float* __restrict__ m_ws,
                             const float* __restrict__ d_ws,
                             float* __restrict__ c_ws, float* __restrict__ A_ws) {
  __shared__ float Bp[NC], msh[NC], dsh[NC];
  int s = blockIdx.x, tid = threadIdx.x;
  msh[tid] = m_ws[s * NC + tid];
  dsh[tid] = d_ws[s * NC + tid];
  __syncthreads();
  if (tid == 0) {
    float p = 1.f;
    for (int i = NC - 1; i >= 0; --i) { Bp[i] = p; p *= (1.f - dsh[i]); }
    A_ws[s] = p;
  }
  __syncthreads();
  float prod = 1.f, c = 0.f;
  for (int i = tid; i < NC; ++i) {
    if (i > tid) prod *= msh[i];
    c += Bp[i] * prod;
  }
  c_ws[s * NC + tid] = c;
}

// Kernel 4: out[s, :] = A_s * params0
__global__ void init_out_kernel(const float* __restrict__ A_ws,
                                const float* __restrict__ nw0,
                                const float* __restrict__ w00,
                                const float* __restrict__ w10,
                                float* __restrict__ out) {
  long long total = (long long)BH * PER_STREAM;
  for (long long idx = (long long)blockIdx.x * 256 + threadIdx.x; idx < total;
       idx += (long long)gridDim.x * 256) {
    int s = (int)(idx / PER_STREAM);
    int r = (int)(idx % PER_STREAM);
    int h = s % HEADS;
    float val;
    if (r < DH) val = nw0[h * DH + r];
    else if (r < DH + DH * DHID) val = w00[(size_t)h * DH * DHID + (r - DH)];
    else val = w10[(size_t)h * DHID * DH + (r - DH - DH * DHID)];
    out[idx] = A_ws[s] * val;
  }
}

// ---------------------------------------------------------------------------
// Kernel 5: per (stream, chunk) MLP fwd+bwd with WMMA f32; atomic accumulate
// -c_{s,j} * grad into out. grid = (64 chunks, 16 streams), 256 threads,
// dyn LDS ~234 KB (one workgroup per WGP; 320 KB available).
// ---------------------------------------------------------------------------
__global__ __launch_bounds__(256) void grad_kernel(
    const float* __restrict__ k_ws, const float* __restrict__ v_ws,
    const float* __restrict__ lr_ws, const float* __restrict__ c_ws,
    const float* __restrict__ nw0, const float* __restrict__ w00,
    const float* __restrict__ w10, float* __restrict__ out) {
  extern __shared__ float sm[];
  float* Ks  = sm;              // [64][66]   k chunk (tokens x dh)
  float* Dp  = Ks + 64 * 66;    // [64][66]   dpred, later dh
  float* W1s = Dp + 64 * 66;    // [256][66]  w1 (hid x dh)
  float* W0s = W1s + 256 * 66;  // [64][258]  w0 (dh x hid)
  float* As  = W0s + 64 * 258;  // [64][258]  preact a, later da
  float* nws = As + 64 * 258;   // [64]
  float* rstd = nws + 64;       // [64]
  float* lrs = rstd + 64;       // [64]

  const int j = blockIdx.x, s = blockIdx.y;
  const int h = s % HEADS;
  const int tid = threadIdx.x, lane = tid & 31, wave = tid >> 5;
  const float negc = -c_ws[s * NC + j];

  const float* kg = k_ws + ((size_t)s * NC + j) * CHUNKT * DH;
  const float* vg = v_ws + ((size_t)s * NC + j) * CHUNKT * DH;
  const float* w0g = w00 + (size_t)h * DH * DHID;
  const float* w1g = w10 + (size_t)h * DHID * DH;

  for (int i = tid; i < 1024; i += 256) {         // k: 64x64
    int r = (i * 4) / 64, c = (i * 4) % 64;
    float4 s4 = *(const float4*)&kg[i * 4];
    float* dst = &Ks[r * 66 + c];
    dst[0] = s4.x; dst[1] = s4.y; dst[2] = s4.z; dst[3] = s4.w;
  }
  for (int i = tid; i < 4096; i += 256) {         // w0: 64x256
    int r = (i * 4) / 256, c = (i * 4) % 256;
    float4 s4 = *(const float4*)&w0g[i * 4];
    float* dst = &W0s[r * 258 + c];
    dst[0] = s4.x; dst[1] = s4.y; dst[2] = s4.z; dst[3] = s4.w;
  }
  for (int i = tid; i < 4096; i += 256) {         // w1: 256x64
    int r = (i * 4) / 64, c = (i * 4) % 64;
    float4 s4 = *(const float4*)&w1g[i * 4];
    float* dst = &W1s[r * 66 + c];
    dst[0] = s4.x; dst[1] = s4.y; dst[2] = s4.z; dst[3] = s4.w;
  }
  if (tid < 64) nws[tid] = nw0[h * DH + tid];
  else if (tid < 128) lrs[tid - 64] = lr_ws[((size_t)s * NC + j) * CHUNKT + (tid - 64)];
  __syncthreads();

  // step 1: rstd per token
  for (int it = 0; it < 8; ++it) {
    int t = wave * 8 + it;
    float x0 = Ks[t * 66 + lane], x1 = Ks[t * 66 + lane + 32];
    float ss = x0 * x0 + x1 * x1;
    for (int o = 16; o > 0; o >>= 1) ss += __shfl_xor(ss, o, 32);
    if (lane == 0) rstd[t] = rsqrtf(ss / DH + EPSV);
  }
  __syncthreads();

  // step 2: a = hhat @ w0  (hhat computed on the fly), 64 tiles
  for (int q = 0; q < 8; ++q) {
    int tI = wave * 8 + q, mi = tI >> 4, ni = tI & 15;
    v8f acc = {};
    int ar = mi * 16 + (lane & 15);
    float rs = rstd[ar];
    int bc = ni * 16 + (lane & 15);
    for (int k4 = 0; k4 < 64; k4 += 4) {
      int kb = k4 + ((lane >> 4) << 1);
      v2f a, bv;
      a.x = Ks[ar * 66 + kb] * rs * nws[kb];
      a.y = Ks[ar * 66 + kb + 1] * rs * nws[kb + 1];
      bv.x = W0s[kb * 258 + bc]; bv.y = W0s[(kb + 1) * 258 + bc];
      acc = wmma4(a, bv, acc);
    }
    for (int i = 0; i < 8; ++i) {
      int r = mi * 16 + i + ((lane >> 4) << 3);
      As[r * 258 + ni * 16 + (lane & 15)] = acc[i];
    }
  }
  __syncthreads();

  // step 3/4: y = gelu(a) @ w1 ; dpred = (2 lr / DH)(y + k - v)
  for (int q = 0; q < 2; ++q) {
    int tI = wave * 2 + q, mi = tI >> 2, ni = tI & 3;
    v8f acc = {};
    int ar = mi * 16 + (lane & 15);
    int bc = ni * 16 + (lane & 15);
    for (int k4 = 0; k4 < 256; k4 += 4) {
      int kb = k4 + ((lane >> 4) << 1);
      v2f a, bv;
      a.x = gelu_f(As[ar * 258 + kb]);
      a.y = gelu_f(As[ar * 258 + kb + 1]);
      bv.x = W1s[kb * 66 + bc]; bv.y = W1s[(kb + 1) * 66 + bc];
      acc = wmma4(a, bv, acc);
    }
    for (int i = 0; i < 8; ++i) {
      int t = mi * 16 + i + ((lane >> 4) << 3);
      int d = ni * 16 + (lane & 15);
      float pred = acc[i] + Ks[t * 66 + d];
      Dp[t * 66 + d] = (2.f / DH) * lrs[t] * (pred - vg[t * DH + d]);
    }
  }
  __syncthreads();

  // step 5: grad_w1 = gelu(a)^T @ dpred -> atomic accumulate
  for (int q = 0; q < 8; ++q) {
    int tI = wave * 8 + q, mi = tI >> 2, ni = tI & 3;
    v8f acc = {};
    int mrow = mi * 16 + (lane & 15);  // hid
    int bc = ni * 16 + (lane & 15);    // d
    for (int k4 = 0; k4 < 64; k4 += 4) {
      int kb = k4 + ((lane >> 4) << 1);  // token
      v2f a, bv;
      a.x = gelu_f(As[kb * 258 + mrow]);
      a.y = gelu_f(As[(kb + 1) * 258 + mrow]);
      bv.x = Dp[kb * 66 + bc]; bv.y = Dp[(kb + 1) * 66 + bc];
      acc = wmma4(a, bv, acc);
    }
    for (int i = 0; i < 8; ++i) {
      int hid = mi * 16 + i + ((lane >> 4) << 3);
      int d = ni * 16 + (lane & 15);
      atomicAdd(&out[(size_t)s * PER_STREAM + DH + DH * DHID + hid * DH + d],
                negc * acc[i]);
    }
  }
  __syncthreads();

  // step 6/7: da = (dpred @ w1^T) * gelu'(a), stored in place of a
  for (int q = 0; q < 8; ++q) {
    int tI = wave * 8 + q, mi = tI >> 4, ni = tI & 15;
    v8f acc = {};
    int ar = mi * 16 + (lane & 15);  // token
    int bn = ni * 16 + (lane & 15);  // hid
    for (int k4 = 0; k4 < 64; k4 += 4) {
      int kb = k4 + ((lane >> 4) << 1);  // d
      v2f a = *(v2f*)&Dp[ar * 66 + kb];
      v2f bv = *(v2f*)&W1s[bn * 66 + kb];
      acc = wmma4(a, bv, acc);
    }
    for (int i = 0; i < 8; ++i) {
      int t = mi * 16 + i + ((lane >> 4) << 3);
      int hid = ni * 16 + (lane & 15);
      float av = As[t * 258 + hid];
      As[t * 258 + hid] = acc[i] * gelu_grad(av);
    }
  }
  __syncthreads();

  // step 8: grad_w0 = hhat^T @ da -> atomic accumulate
  for (int q = 0; q < 8; ++q) {
    int tI = wave * 8 + q, mi = tI >> 4, ni = tI & 15;
    v8f acc = {};
    int kd = mi * 16 + (lane & 15);
    int bc = ni * 16 + (lane & 15);  // hid
    float nwv = nws[kd];
    for (int k4 = 0; k4 < 64; k4 += 4) {
      int kb = k4 + ((lane >> 4) << 1);  // token
      v2f a, bv;
      a.x = Ks[kb * 66 + kd] * rstd[kb] * nwv;
      a.y = Ks[(kb + 1) * 66 + kd] * rstd[kb + 1] * nwv;
      bv.x = As[kb * 258 + bc]; bv.y = As[(kb + 1) * 258 + bc];
      acc = wmma4(a, bv, acc);
    }
    for (int i = 0; i < 8; ++i) {
      int r = mi * 16 + i + ((lane >> 4) << 3);
      int c = ni * 16 + (lane & 15);
      atomicAdd(&out[(size_t)s * PER_STREAM + DH + r * DHID + c], negc * acc[i]);
    }
  }

  // step 9: dh = da @ w0^T -> Dp
  for (int q = 0; q < 2; ++q) {
    int tI = wave * 2 + q, mi = tI >> 2, ni = tI & 3;
    v8f acc = {};
    int ar = mi * 16 + (lane & 15);  // token
    int bn = ni * 16 + (lane & 15);  // kd
    for (int k4 = 0; k4 < 256; k4 += 4) {
      int kb = k4 + ((lane >> 4) << 1);  // hid
      v2f a = *(v2f*)&As[ar * 258 + kb];
      v2f bv = *(v2f*)&W0s[bn * 258 + kb];
      acc = wmma4(a, bv, acc);
    }
    for (int i = 0; i < 8; ++i) {
      int t = mi * 16 + i + ((lane >> 4) << 3);
      Dp[t * 66 + ni * 16 + (lane & 15)] = acc[i];
    }
  }
  __syncthreads();

  // step 10: grad_nw[d] = sum_t dh[t][d] * k[t][d] * rstd[t]
  if (tid < 64) {
    float g = 0.f;
    for (int t = 0; t < 64; ++t) g += Dp[t * 66 + tid] * Ks[t * 66 + tid] * rstd[t];
    atomicAdd(&out[(size_t)s * PER_STREAM + tid], negc * g);
  }
}

extern "C" void kernel_launch(void* const* d_in, const int* in_sizes, int n_in,
                              void* d_out, int out_size, void* d_ws, size_t ws_size,
                              hipStream_t stream) {
  (void)in_sizes; (void)n_in; (void)out_size; (void)ws_size;
  const float* seq   = (const float*)d_in[0];
  const float* snw   = (const float*)d_in[1];
  const float* Wk    = (const float*)d_in[2];
  const float* Wv    = (const float*)d_in[3];
  const float* Wstep = (const float*)d_in[4];
  const float* bstep = (const float*)d_in[5];
  const float* Wmom  = (const float*)d_in[6];
  const float* bmom  = (const float*)d_in[7];
  const float* Wdec  = (const float*)d_in[8];
  const float* bdec  = (const float*)d_in[9];
  const float* nw0   = (const float*)d_in[10];
  const float* w00   = (const float*)d_in[11];
  const float* w10   = (const float*)d_in[12];
  float* out = (float*)d_out;
  float* ws  = (float*)d_ws;

  float* xhat  = ws;                                   // 2*4096*512
  float* k_ws  = xhat + (size_t)2 * 4096 * 512;        // 16*64*64*64
  float* v_ws  = k_ws + (size_t)BH * NC * CHUNKT * DH;
  float* lr_ws = v_ws + (size_t)BH * NC * CHUNKT * DH; // 16*64*64
  float* m_ws  = lr_ws + (size_t)BH * NC * CHUNKT;
  float* d_wsp = m_ws + BH * NC;
  float* c_ws  = d_wsp + BH * NC;
  float* A_ws  = c_ws + BH * NC;

  size_t sm1 = (size_t)(CHUNKT * DIMX + DIMX) * sizeof(float);
  prep_kernel<<<dim3(NB * NC), dim3(256), sm1, stream>>>(
      seq, snw, Wstep, bstep, Wmom, bmom, Wdec, bdec, xhat, lr_ws, m_ws, d_wsp);

  size_t sm2 = (size_t)(2 * 64 * 66) * sizeof(float);
  kv_gemm_kernel<<<dim3(NB * NC, HEADS, 2), dim3(256), sm2, stream>>>(
      xhat, Wk, Wv, k_ws, v_ws);

  coeff_kernel<<<dim3(BH), dim3(NC), 0, stream>>>(m_ws, d_wsp, c_ws, A_ws);
  init_out_kernel<<<dim3(512), dim3(256), 0, stream>>>(A_ws, nw0, w00, w10, out);

  size_t sm5 = (size_t)(64 * 66 * 2 + 256 * 66 + 64 * 258 * 2 + 192) * sizeof(float);
  grad_kernel<<<dim3(NC, BH), dim3(256), sm5, stream>>>(
      k_ws, v_ws, lr_ws, c_ws, nw0, w00, w10, out);
}